// GnnProjector_67482526155352
// MI455X (gfx1250) — compile-verified
//
#include <hip/hip_runtime.h>
#include <hip/hip_bf16.h>

typedef float v2f __attribute__((ext_vector_type(2)));
typedef float v8f __attribute__((ext_vector_type(8)));

#define HID 128
#define PROJ 256
#define NLAYER 5
#define NAF 9
#define NBF 3
#define BVOC 16
#define BN_EPS 1e-5f

// ---------------- Atom encoder: h[n,:] = sum_f atom_emb[f, x[n,f], :] ----------------
// one wave (32 lanes) per node, float4 per lane covers 128 channels
__global__ void atom_enc_kernel(const int* __restrict__ x,
                                const float* __restrict__ atom_emb,
                                float* __restrict__ h, int N) {
    int wid = (blockIdx.x * blockDim.x + threadIdx.x) >> 5;
    if (wid >= N) return;
    int lane = threadIdx.x & 31;
    float4 acc = make_float4(0.f, 0.f, 0.f, 0.f);
#pragma unroll
    for (int f = 0; f < NAF; ++f) {
        int idx = x[wid * NAF + f];
        const float4* row = (const float4*)(atom_emb + ((size_t)f * 128 + idx) * HID);
        float4 v = row[lane];
        acc.x += v.x; acc.y += v.y; acc.z += v.z; acc.w += v.w;
    }
    ((float4*)(h + (size_t)wid * HID))[lane] = acc;
}

// ---------------- z0 = (1+eps[l]) * h ----------------
__global__ void init_z_kernel(const float* __restrict__ h, const float* __restrict__ eps,
                              int l, float* __restrict__ z0, int total4) {
    int i = blockIdx.x * blockDim.x + threadIdx.x;
    if (i >= total4) return;
    float s = 1.0f + eps[l];
    float4 v = ((const float4*)h)[i];
    v.x *= s; v.y *= s; v.z *= s; v.w *= s;
    ((float4*)z0)[i] = v;
}

// ---------------- edge message + scatter-add ----------------
// one wave per edge: msg = relu(h[src] + bond_emb lookups), atomic-add into z0[dst]
__global__ void edge_msg_kernel(const float* __restrict__ h,
                                const int* __restrict__ edge_index,
                                const int* __restrict__ edge_attr,
                                const float* __restrict__ bond_emb_l,
                                float* __restrict__ z0, int E) {
    int wid = (blockIdx.x * blockDim.x + threadIdx.x) >> 5;
    if (wid >= E) return;
    int lane = threadIdx.x & 31;
    int src = edge_index[wid];
    int dst = edge_index[E + wid];
    int a0 = edge_attr[wid * NBF + 0];
    int a1 = edge_attr[wid * NBF + 1];
    int a2 = edge_attr[wid * NBF + 2];
    const float4* hr = (const float4*)(h + (size_t)src * HID);
    const float4* e0 = (const float4*)(bond_emb_l + ((size_t)0 * BVOC + a0) * HID);
    const float4* e1 = (const float4*)(bond_emb_l + ((size_t)1 * BVOC + a1) * HID);
    const float4* e2 = (const float4*)(bond_emb_l + ((size_t)2 * BVOC + a2) * HID);
    float4 v = hr[lane], w0 = e0[lane], w1 = e1[lane], w2 = e2[lane];
    float m0 = fmaxf(v.x + w0.x + w1.x + w2.x, 0.f);
    float m1 = fmaxf(v.y + w0.y + w1.y + w2.y, 0.f);
    float m2 = fmaxf(v.z + w0.z + w1.z + w2.z, 0.f);
    float m3 = fmaxf(v.w + w0.w + w1.w + w2.w, 0.f);
    float* zr = z0 + (size_t)dst * HID + lane * 4;
    atomicAdd(zr + 0, m0);
    atomicAdd(zr + 1, m1);
    atomicAdd(zr + 2, m2);
    atomicAdd(zr + 3, m3);
}

// ---------------- WMMA f32 GEMM: C[M,NC] = op(A)[M,K] @ B[K,NC] + bias ----------------
// Block = 8 waves covering a 128-row x 32-col region. B[K x 32] is staged once into
// LDS in (k,k+1)-pair layout so each lane's B fragment is a single ds_load_b64.
// Each wave computes a 16x32 tile: one A fragment feeds two V_WMMA_F32_16X16X4_F32.
// BN_RELU_A: apply per-k scale/shift + relu to A elements on load (fused BN1+ReLU).
template <bool BN_RELU_A>
__global__ __launch_bounds__(256) void gemm_wmma_kernel(
    const float* __restrict__ A, const float* __restrict__ B,
    const float* __restrict__ bias,
    const float* __restrict__ a_scale, const float* __restrict__ a_shift,
    float* __restrict__ C, int M, int K, int NC) {
    extern __shared__ float lds_b[];  // [2][K/2][16 pairs]  (t, k>>1, n*2 + (k&1))
    int tid = threadIdx.x;
    int n0 = blockIdx.y << 5;  // 32-column span

    // stage B[K x 32] -> LDS (coalesced global reads, pair-swizzled LDS writes)
    int total = K << 5;
    for (int idx = tid; idx < total; idx += 256) {
        int k = idx >> 5;
        int nn = idx & 31;
        float v = B[(size_t)k * NC + n0 + nn];
        int t = nn >> 4, n = nn & 15;
        lds_b[t * (K << 4) + ((k >> 1) << 5) + n * 2 + (k & 1)] = v;
    }
    __syncthreads();

    int wave = tid >> 5;
    int mt = blockIdx.x * 8 + wave;
    int lane = tid & 31;
    int half = lane >> 4;  // 0: K pair {0,1}, 1: K pair {2,3}
    int sub = lane & 15;   // row (A) / col (B,C)
    bool active = mt < (M >> 4);
    int m0 = mt << 4;

    if (active) {
        v8f c0 = {}, c1 = {};
        const float* arow = A + (size_t)(m0 + sub) * K;
        const float* lb0 = lds_b + half * 32 + sub * 2;
        const float* lb1 = lb0 + (K << 4);

#pragma unroll 4
        for (int k0 = 0; k0 < K; k0 += 4) {
            int ka = k0 + half * 2;
            v2f a;
            if (BN_RELU_A) {
                a.x = fmaxf(arow[ka + 0] * a_scale[ka + 0] + a_shift[ka + 0], 0.f);
                a.y = fmaxf(arow[ka + 1] * a_scale[ka + 1] + a_shift[ka + 1], 0.f);
            } else {
                v2f tmp = *(const v2f*)(arow + ka);
                a = tmp;
            }
            v2f b0 = *(const v2f*)(lb0 + ((k0 >> 1) << 5));
            v2f b1 = *(const v2f*)(lb1 + ((k0 >> 1) << 5));
            c0 = __builtin_amdgcn_wmma_f32_16x16x4_f32(false, a, false, b0, (short)0,
                                                       c0, false, false);
            c1 = __builtin_amdgcn_wmma_f32_16x16x4_f32(false, a, false, b1, (short)0,
                                                       c1, false, false);
        }

        float bv0 = bias[n0 + sub];
        float bv1 = bias[n0 + 16 + sub];
#pragma unroll
        for (int j = 0; j < 8; ++j) {
            int row = m0 + j + half * 8;
            C[(size_t)row * NC + n0 + sub] = c0[j] + bv0;
            C[(size_t)row * NC + n0 + 16 + sub] = c1[j] + bv1;
        }
    }
}

// ---------------- BN statistics ----------------
__global__ void zero_stats_kernel(float* __restrict__ sum, float* __restrict__ sq, int C) {
    int i = threadIdx.x;
    if (i < C) { sum[i] = 0.f; sq[i] = 0.f; }
}

// blockDim.x == C; coalesced row-chunk accumulation, 2 atomics per (block,col)
__global__ void stats_partial_kernel(const float* __restrict__ Z,
                                     float* __restrict__ sum, float* __restrict__ sq,
                                     int M, int C) {
    int col = threadIdx.x;
    int rpb = (M + gridDim.x - 1) / gridDim.x;
    int r0 = blockIdx.x * rpb;
    int r1 = min(M, r0 + rpb);
    float s = 0.f, q = 0.f;
    for (int r = r0; r < r1; ++r) {
        float v = Z[(size_t)r * C + col];
        s += v; q += v * v;
    }
    atomicAdd(&sum[col], s);
    atomicAdd(&sq[col], q);
}

__global__ void stats_finalize_kernel(const float* __restrict__ sum,
                                      const float* __restrict__ sq,
                                      const float* __restrict__ gamma,
                                      const float* __restrict__ beta,
                                      float* __restrict__ scale,
                                      float* __restrict__ shift, int M, int C) {
    int c = threadIdx.x;
    if (c >= C) return;
    float inv = 1.0f / (float)M;
    float mean = sum[c] * inv;
    float var  = sq[c] * inv - mean * mean;
    float rstd = rsqrtf(var + BN_EPS);
    float sc = gamma[c] * rstd;
    scale[c] = sc;
    shift[c] = beta[c] - mean * sc;
}

// ---------------- elementwise BN apply (+ optional relu) ----------------
__global__ void bn_apply_kernel(const float* __restrict__ Z,
                                const float* __restrict__ scale,
                                const float* __restrict__ shift,
                                float* __restrict__ H, int total, int C, int do_relu) {
    int i = blockIdx.x * blockDim.x + threadIdx.x;
    if (i >= total) return;
    int c = i & (C - 1);
    float v = Z[i] * scale[c] + shift[c];
    if (do_relu) v = fmaxf(v, 0.f);
    H[i] = v;
}

extern "C" void kernel_launch(void* const* d_in, const int* in_sizes, int n_in,
                              void* d_out, int out_size, void* d_ws, size_t ws_size,
                              hipStream_t stream) {
    const int*   x          = (const int*)d_in[0];
    const int*   edge_index = (const int*)d_in[1];
    const int*   edge_attr  = (const int*)d_in[2];
    const float* atom_emb   = (const float*)d_in[3];
    const float* bond_emb   = (const float*)d_in[4];
    const float* eps        = (const float*)d_in[5];
    const float* w1         = (const float*)d_in[6];
    const float* b1         = (const float*)d_in[7];
    const float* bn1_g      = (const float*)d_in[8];
    const float* bn1_b      = (const float*)d_in[9];
    const float* w2         = (const float*)d_in[10];
    const float* b2         = (const float*)d_in[11];
    const float* bn_g       = (const float*)d_in[12];
    const float* bn_b       = (const float*)d_in[13];
    const float* lin_w      = (const float*)d_in[14];
    const float* lin_b      = (const float*)d_in[15];

    const int N = in_sizes[0] / NAF;
    const int E = in_sizes[1] / 2;

    // workspace layout (floats)
    float* h     = (float*)d_ws;
    float* z0    = h  + (size_t)N * HID;
    float* z2    = z0 + (size_t)N * HID;
    float* s_sum = z2 + (size_t)N * HID;
    float* s_sq    = s_sum + PROJ;
    float* s_scale = s_sq + PROJ;
    float* s_shift = s_scale + PROJ;
    float* z1 = (float*)d_out;  // reuse output buffer as [N,256] scratch

    const int BLK = 256;
    const int Mtiles = N / 16;
    const int gx = (Mtiles + 7) / 8;  // 8 m-tiles per block

    // atom encoder (wave per node)
    atom_enc_kernel<<<(N * 32 + BLK - 1) / BLK, BLK, 0, stream>>>(x, atom_emb, h, N);

    for (int l = 0; l < NLAYER; ++l) {
        const float* bond_l = bond_emb + (size_t)l * NBF * BVOC * HID;
        const float* w1l = w1 + (size_t)l * HID * 2 * HID;
        const float* b1l = b1 + (size_t)l * 2 * HID;
        const float* w2l = w2 + (size_t)l * 2 * HID * HID;
        const float* b2l = b2 + (size_t)l * HID;

        // z0 = (1+eps)*h
        int tot4 = N * HID / 4;
        init_z_kernel<<<(tot4 + BLK - 1) / BLK, BLK, 0, stream>>>(h, eps, l, z0, tot4);

        // scatter-add relu(h[src]+e) into z0[dst]  (wave per edge)
        edge_msg_kernel<<<(E * 32 + BLK - 1) / BLK, BLK, 0, stream>>>(
            h, edge_index, edge_attr, bond_l, z0, E);

        // GEMM1: z1[N,256] = z0 @ w1 + b1   (K=128, NC=256)
        gemm_wmma_kernel<false><<<dim3(gx, (2 * HID) / 32), BLK,
                                  (size_t)HID * 32 * sizeof(float), stream>>>(
            z0, w1l, b1l, nullptr, nullptr, z1, N, HID, 2 * HID);

        // BN1 stats over z1 columns (C=256)
        zero_stats_kernel<<<1, PROJ, 0, stream>>>(s_sum, s_sq, 2 * HID);
        stats_partial_kernel<<<128, 2 * HID, 0, stream>>>(z1, s_sum, s_sq, N, 2 * HID);
        stats_finalize_kernel<<<1, PROJ, 0, stream>>>(
            s_sum, s_sq, bn1_g + (size_t)l * 2 * HID, bn1_b + (size_t)l * 2 * HID,
            s_scale, s_shift, N, 2 * HID);

        // GEMM2 with fused BN1+ReLU on A: z2[N,128] = relu(bn1(z1)) @ w2 + b2
        gemm_wmma_kernel<true><<<dim3(gx, HID / 32), BLK,
                                 (size_t)(2 * HID) * 32 * sizeof(float), stream>>>(
            z1, w2l, b2l, s_scale, s_shift, z2, N, 2 * HID, HID);

        // outer BN stats over z2 columns (C=128)
        zero_stats_kernel<<<1, PROJ, 0, stream>>>(s_sum, s_sq, HID);
        stats_partial_kernel<<<128, HID, 0, stream>>>(z2, s_sum, s_sq, N, HID);
        stats_finalize_kernel<<<1, PROJ, 0, stream>>>(
            s_sum, s_sq, bn_g + (size_t)l * HID, bn_b + (size_t)l * HID,
            s_scale, s_shift, N, HID);

        // h = bn(z2) (+relu except last layer)
        int tot = N * HID;
        bn_apply_kernel<<<(tot + BLK - 1) / BLK, BLK, 0, stream>>>(
            z2, s_scale, s_shift, h, tot, HID, (l < NLAYER - 1) ? 1 : 0);
    }

    // final projection: out = h @ lin_w + lin_b
    gemm_wmma_kernel<false><<<dim3(gx, PROJ / 32), BLK,
                              (size_t)HID * 32 * sizeof(float), stream>>>(
        h, lin_w, lin_b, nullptr, nullptr, (float*)d_out, N, HID, PROJ);
}